// IndustrySparseAttention_31920196944042
// MI455X (gfx1250) — compile-verified
//
#include <hip/hip_runtime.h>
#include <math.h>

// ---------------------------------------------------------------------------
// IndustrySparseAttention for MI455X (gfx1250, wave32)
//   B=2, S=4096, E=128, H=4, D=32, NUM_GROUPS=64
//   - group-bucketed (block-sparse) flash attention, fp32 WMMA 16x16x4
//   - GEMMs: LDS-staged X and transposed-W, pure ds_load_b64 + v_wmma inner loop
//   - attention K/V tiles staged via GLOBAL_LOAD_ASYNC_TO_LDS_B128 (ASYNCcnt)
// ---------------------------------------------------------------------------

typedef __attribute__((ext_vector_type(2))) float v2f;
typedef __attribute__((ext_vector_type(8))) float v8f;

#define SEQ   4096
#define EMB   128
#define HEADS 4
#define HDIM  32
#define NGRP  64
#define MAXPG 160     // cap on tokens per group (mean 64, sigma ~8 -> safe)
#define MAXTILES 10   // ceil(MAXPG/16)

#define QK_STR 36     // LDS row stride for 32-wide Q/K/V tiles (16B aligned, conflict-free)
#define P_STR  20     // LDS row stride for 16-wide P tile (8B aligned, conflict-free)

// 16-lane butterfly reductions (a C-fragment row lives in 16 lanes of a half)
__device__ __forceinline__ float rowmax16(float v) {
  v = fmaxf(v, __shfl_xor(v, 1, 16));
  v = fmaxf(v, __shfl_xor(v, 2, 16));
  v = fmaxf(v, __shfl_xor(v, 4, 16));
  v = fmaxf(v, __shfl_xor(v, 8, 16));
  return v;
}
__device__ __forceinline__ float rowsum16(float v) {
  v += __shfl_xor(v, 1, 16);
  v += __shfl_xor(v, 2, 16);
  v += __shfl_xor(v, 4, 16);
  v += __shfl_xor(v, 8, 16);
  return v;
}

// Async copy of 64 contiguous bytes global -> LDS (per lane), ASYNCcnt-tracked.
// INST_OFFSET is added to BOTH the LDS and global addresses (ISA 08 §4.4),
// so one (lds,global) base pair covers the whole 64B segment.
__device__ __forceinline__ void async_copy64(unsigned lds_addr, const float* gsrc) {
  asm volatile(
      "global_load_async_to_lds_b128 %0, %1, off\n\t"
      "global_load_async_to_lds_b128 %0, %1, off offset:16\n\t"
      "global_load_async_to_lds_b128 %0, %1, off offset:32\n\t"
      "global_load_async_to_lds_b128 %0, %1, off offset:48"
      :: "v"(lds_addr), "v"(gsrc) : "memory");
}
__device__ __forceinline__ void wait_async0() {
  asm volatile("s_wait_asynccnt 0x0" ::: "memory");
}

// ---------------------------------------------------------------------------
// Y[m, 0:128] = (X[m, 0:128] @ W[128,128] + bias) * scale
// 256-thread block = 8 waves; block computes a 64-row stripe.
// W staged transposed in LDS ([n][k], stride 34): B operand = aligned float2,
// conflict-free. X staged with stride 132. Inner loop: 2x ds_load_b64 + wmma.
// ---------------------------------------------------------------------------
__global__ __launch_bounds__(256)
void gemm128(const float* __restrict__ X, const float* __restrict__ W,
             const float* __restrict__ bias, float* __restrict__ Y,
             float scale) {
  __shared__ float ldsX[64 * 132];    // 33792 B
  __shared__ float ldsWt[128 * 34];   // 17408 B, transposed W: [n][k]
  const int m0 = blockIdx.x * 64;
  const int t  = threadIdx.x;

  __builtin_prefetch(W, 0, 1);        // global_prefetch_b8

#pragma unroll
  for (int i = 0; i < 32; ++i) {      // 64 rows x 128 cols of X, coalesced
    const int lin = i * 256 + t;
    const int row = lin >> 7, col = lin & 127;
    ldsX[row * 132 + col] = X[(long)(m0 + row) * EMB + col];
  }
#pragma unroll
  for (int i = 0; i < 64; ++i) {      // all of W, stored transposed
    const int lin = i * 256 + t;
    const int k = lin >> 7, nn = lin & 127;
    ldsWt[nn * 34 + k] = W[lin];
  }
  __syncthreads();

  const int wave = t >> 5;
  const int lane = t & 31;
  const int nl = lane & 15;           // A row / B,C,D column within tile
  const int hi = lane >> 4;           // K-half selector
  const int rbase = (wave >> 1) * 16; // row tile for this wave (0..3)

#pragma unroll
  for (int nt = 0; nt < 4; ++nt) {
    const int n0 = ((wave & 1) * 4 + nt) * 16;
    v8f c = {};
#pragma unroll
    for (int kc = 0; kc < 32; ++kc) {
      const int k0 = kc * 4 + hi * 2;
      const v2f av = *(const v2f*)&ldsX[(rbase + nl) * 132 + k0];
      const v2f bv = *(const v2f*)&ldsWt[(n0 + nl) * 34 + k0];
      c = __builtin_amdgcn_wmma_f32_16x16x4_f32(false, av, false, bv,
                                                (short)0, c, false, false);
    }
    const float bb = bias[n0 + nl];
#pragma unroll
    for (int i = 0; i < 8; ++i)
      Y[(long)(m0 + rbase + i + 8 * hi) * EMB + n0 + nl] = (c[i] + bb) * scale;
  }
}

// ---------------------------------------------------------------------------
// Deterministic per-group token bucketing (order-preserving).
// ---------------------------------------------------------------------------
__global__ __launch_bounds__(64)
void build_groups(const int* __restrict__ gid, int* __restrict__ lists,
                  int* __restrict__ counts) {
  const int g = threadIdx.x;   // 64 threads, one per group
  int cnt = 0;
  for (int s = 0; s < SEQ; ++s) {
    if (gid[s] == g) {
      if (cnt < MAXPG) lists[g * MAXPG + cnt] = s;
      ++cnt;
    }
  }
  counts[g] = cnt > MAXPG ? MAXPG : cnt;
}

// ---------------------------------------------------------------------------
// Intra-group flash attention. One wave per (b, h, group, 16-query tile).
// Q already carries 1/sqrt(D). K/V tiles arrive via async global->LDS DMA.
// ---------------------------------------------------------------------------
__global__ __launch_bounds__(32)
void attn_kernel(const float* __restrict__ Q, const float* __restrict__ K,
                 const float* __restrict__ V, const int* __restrict__ lists,
                 const int* __restrict__ counts, float* __restrict__ AO) {
  __shared__ float ldsQ[16 * QK_STR];
  __shared__ float ldsK[16 * QK_STR];
  __shared__ float ldsV[16 * QK_STR];
  __shared__ float ldsP[16 * P_STR];

  const int g  = blockIdx.x / MAXTILES;
  const int qt = blockIdx.x % MAXTILES;
  const int n  = counts[g];
  if (qt * 16 >= n) return;                 // uniform exit: EXEC stays all-ones

  const int b = blockIdx.y >> 2;
  const int h = blockIdx.y & 3;
  const int lane = threadIdx.x;
  const int nl = lane & 15;
  const int hi = lane >> 4;
  const int* gl = lists + g * MAXPG;
  const long base = (long)b * SEQ * EMB;

  const int r = lane >> 1, half = lane & 1;   // staging role: row r, 16-col half

  // stage the 16x32 Q tile (rows clamped), vectorized float4
  {
    int qr = qt * 16 + r; if (qr > n - 1) qr = n - 1;
    const int tok = gl[qr];
    const float4* src =
        (const float4*)(Q + base + (long)tok * EMB + h * HDIM + half * 16);
    float4* dst = (float4*)(ldsQ + r * QK_STR + half * 16);
#pragma unroll
    for (int c = 0; c < 4; ++c) dst[c] = src[c];
  }

  v8f o0 = {}, o1 = {};
  float m_i[8], l_i[8];
#pragma unroll
  for (int i = 0; i < 8; ++i) { m_i[i] = -INFINITY; l_i[i] = 0.f; }

  const int ktiles = (n + 15) >> 4;
  for (int kt = 0; kt < ktiles; ++kt) {
    // stage 16x32 K and V tiles via async DMA (64B per lane per matrix)
    {
      int kr = kt * 16 + r; if (kr > n - 1) kr = n - 1;
      const int tok = gl[kr];
      const float* ks = K + base + (long)tok * EMB + h * HDIM + half * 16;
      const float* vs = V + base + (long)tok * EMB + h * HDIM + half * 16;
      async_copy64((unsigned)(uintptr_t)(ldsK + r * QK_STR + half * 16), ks);
      async_copy64((unsigned)(uintptr_t)(ldsV + r * QK_STR + half * 16), vs);
      wait_async0();
    }

    // S = Q * K^T  (16x16 tile over D=32 -> 8 WMMAs, K=4 each)
    v8f s = {};
#pragma unroll
    for (int kc = 0; kc < 8; ++kc) {
      const int k0 = kc * 4 + hi * 2;
      const v2f av = *(const v2f*)&ldsQ[nl * QK_STR + k0];
      const v2f bv = *(const v2f*)&ldsK[nl * QK_STR + k0];
      s = __builtin_amdgcn_wmma_f32_16x16x4_f32(false, av, false, bv,
                                                (short)0, s, false, false);
    }

    const bool colvalid = (kt * 16 + nl) < n;

    // online softmax per row (row i+8*hi spans the 16 lanes of this half)
#pragma unroll
    for (int i = 0; i < 8; ++i) {
      const float sv = colvalid ? s[i] : -INFINITY;
      const float mt = rowmax16(sv);          // finite: >=1 valid key per tile
      const float mnew = fmaxf(m_i[i], mt);
      const float p = __expf(sv - mnew);      // exp(-inf) = 0 for masked cols
      const float alpha = __expf(m_i[i] - mnew);
      l_i[i] = l_i[i] * alpha + rowsum16(p);
      m_i[i] = mnew;
      o0[i] *= alpha;
      o1[i] *= alpha;
      ldsP[(i + 8 * hi) * P_STR + nl] = p;    // C-layout -> LDS (A-layout src)
    }

    // O += P * V  (K=16 keys -> 4 WMMAs per 16-col half of D)
#pragma unroll
    for (int kc = 0; kc < 4; ++kc) {
      const int k0 = kc * 4 + hi * 2;
      const v2f av = *(const v2f*)&ldsP[nl * P_STR + k0];
      v2f b0, b1;
      b0.x = ldsV[k0 * QK_STR + nl];
      b0.y = ldsV[(k0 + 1) * QK_STR + nl];
      b1.x = ldsV[k0 * QK_STR + 16 + nl];
      b1.y = ldsV[(k0 + 1) * QK_STR + 16 + nl];
      o0 = __builtin_amdgcn_wmma_f32_16x16x4_f32(false, av, false, b0,
                                                 (short)0, o0, false, false);
      o1 = __builtin_amdgcn_wmma_f32_16x16x4_f32(false, av, false, b1,
                                                 (short)0, o1, false, false);
    }
  }

  // normalize and scatter rows back to [B, S, E] head slice
#pragma unroll
  for (int i = 0; i < 8; ++i) {
    const int rr = qt * 16 + i + 8 * hi;
    if (rr < n) {
      const int tok = gl[rr];
      const float inv = 1.0f / l_i[i];
      float* dst = AO + base + (long)tok * EMB + h * HDIM;
      dst[nl]      = o0[i] * inv;
      dst[16 + nl] = o1[i] * inv;
    }
  }
}

// ---------------------------------------------------------------------------
// launch — inputs: x, group_ids, Wq, bq, Wk, bk, Wv, bv, Wo, bo
// ---------------------------------------------------------------------------
extern "C" void kernel_launch(void* const* d_in, const int* in_sizes, int n_in,
                              void* d_out, int out_size, void* d_ws, size_t ws_size,
                              hipStream_t stream) {
  const float* x   = (const float*)d_in[0];
  const int*   gid = (const int*)d_in[1];
  const float* Wq  = (const float*)d_in[2];
  const float* bq  = (const float*)d_in[3];
  const float* Wk  = (const float*)d_in[4];
  const float* bk  = (const float*)d_in[5];
  const float* Wv  = (const float*)d_in[6];
  const float* bv  = (const float*)d_in[7];
  const float* Wo  = (const float*)d_in[8];
  const float* bo  = (const float*)d_in[9];
  float* out = (float*)d_out;

  const long NTOK = 2L * SEQ;            // 8192 rows
  const long MAT  = NTOK * EMB;          // 1,048,576 floats per matrix
  float* ws = (float*)d_ws;
  float* Qb = ws;
  float* Kb = ws + MAT;
  float* Vb = ws + 2 * MAT;
  float* AO = ws + 3 * MAT;
  int* lists  = (int*)(ws + 4 * MAT);
  int* counts = lists + NGRP * MAXPG;

  const float qscale = 0.1767766952966369f;   // 1/sqrt(HDIM)

  const int gblocks = (int)(NTOK / 64);       // 128 blocks of 256 threads
  gemm128<<<gblocks, 256, 0, stream>>>(x, Wq, bq, Qb, qscale);
  gemm128<<<gblocks, 256, 0, stream>>>(x, Wk, bk, Kb, 1.0f);
  gemm128<<<gblocks, 256, 0, stream>>>(x, Wv, bv, Vb, 1.0f);

  build_groups<<<1, 64, 0, stream>>>(gid, lists, counts);

  attn_kernel<<<dim3(NGRP * MAXTILES, 2 * HEADS), 32, 0, stream>>>(
      Qb, Kb, Vb, lists, counts, AO);

  gemm128<<<gblocks, 256, 0, stream>>>(AO, Wo, bo, out, 1.0f);
}